// BernNet_86492051407432
// MI455X (gfx1250) — compile-verified
//
#include <hip/hip_runtime.h>
#include <hip/hip_bf16.h>
#include <math.h>

#define N_NODES 100000
#define N_EDGES 1600000
#define F_IN    500
#define HID     64
#define CLS     64
#define K_ORDER 10

typedef float v2f __attribute__((ext_vector_type(2)));
typedef float v8f __attribute__((ext_vector_type(8)));

// ---------------------------------------------------------------------------
// Fused MLP: h = relu(x@W1+b1)@W2+b2 using V_WMMA_F32_16X16X4_F32.
// Block = 256 threads = 8 waves; each wave owns a 16-row tile (block: 128 rows).
// A-fragment layout (16x4 f32): lane L -> M = L%16, VGPR v holds K = v + 2*(L/16).
// B-fragment layout (4x16 f32): lane L -> N = L%16, VGPR v holds K = v + 2*(L/16).
// C/D layout (16x16 f32): VGPR v, lane L -> M = v + 8*(L/16), N = L%16.
// ---------------------------------------------------------------------------
__global__ __launch_bounds__(256)
void mlp_wmma_kernel(const float* __restrict__ x,
                     const float* __restrict__ W1,
                     const float* __restrict__ b1,
                     const float* __restrict__ W2,
                     const float* __restrict__ b2,
                     float* __restrict__ h)
{
    __shared__ float lds_h[8][16 * 68];   // per-wave 16x64 tile, stride 68 (bank-pad)

    const int lane = threadIdx.x & 31;
    const int wave = threadIdx.x >> 5;
    const int half = lane >> 4;           // 0 or 1
    const int l16  = lane & 15;
    const int rowbase = blockIdx.x * 128 + wave * 16;

    // Clamp out-of-range rows (EXEC must stay all-ones for WMMA); mask on store.
    int arow = rowbase + l16;
    if (arow >= N_NODES) arow = N_NODES - 1;
    const float* xrow = x + (size_t)arow * F_IN;

    // ---- Stage 1: [16 x 500] @ [500 x 64] -----------------------------------
    v8f acc[4] = {};
    for (int k0 = 0; k0 < F_IN; k0 += 4) {
        v2f a;
        a.x = xrow[k0 + 2 * half];
        a.y = xrow[k0 + 2 * half + 1];
#pragma unroll
        for (int t = 0; t < 4; ++t) {
            const int colN = t * 16 + l16;
            v2f b;
            b.x = W1[(size_t)(k0 + 2 * half)     * HID + colN];
            b.y = W1[(size_t)(k0 + 2 * half + 1) * HID + colN];
            acc[t] = __builtin_amdgcn_wmma_f32_16x16x4_f32(
                false, a, false, b, (short)0, acc[t], false, false);
        }
    }

    // bias + relu, stage tile to LDS row-major for stage-2 A-fragments
#pragma unroll
    for (int t = 0; t < 4; ++t) {
        const int colN = t * 16 + l16;
        const float bias = b1[colN];
#pragma unroll
        for (int v = 0; v < 8; ++v) {
            const int m = v + 8 * half;
            float val = acc[t][v] + bias;
            lds_h[wave][m * 68 + colN] = val > 0.0f ? val : 0.0f;
        }
    }
    __syncthreads();

    // ---- Stage 2: [16 x 64] @ [64 x 64] -------------------------------------
    v8f acc2[4] = {};
    for (int k0 = 0; k0 < HID; k0 += 4) {
        v2f a;
        a.x = lds_h[wave][l16 * 68 + k0 + 2 * half];
        a.y = lds_h[wave][l16 * 68 + k0 + 2 * half + 1];
#pragma unroll
        for (int t = 0; t < 4; ++t) {
            const int colN = t * 16 + l16;
            v2f b;
            b.x = W2[(size_t)(k0 + 2 * half)     * CLS + colN];
            b.y = W2[(size_t)(k0 + 2 * half + 1) * CLS + colN];
            acc2[t] = __builtin_amdgcn_wmma_f32_16x16x4_f32(
                false, a, false, b, (short)0, acc2[t], false, false);
        }
    }

#pragma unroll
    for (int t = 0; t < 4; ++t) {
        const int colN = t * 16 + l16;
        const float bias = b2[colN];
#pragma unroll
        for (int v = 0; v < 8; ++v) {
            const int grow = rowbase + v + 8 * half;
            if (grow < N_NODES)
                h[(size_t)grow * CLS + colN] = acc2[t][v] + bias;
        }
    }
}

// ---------------------------------------------------------------------------
// Graph preprocessing: degrees, normalized weights, CSR-by-destination build
// ---------------------------------------------------------------------------
__global__ void izero_kernel(int* __restrict__ p, int n) {
    int i = blockIdx.x * blockDim.x + threadIdx.x;
    if (i < n) p[i] = 0;
}

__global__ void count_kernel(const int* __restrict__ row, int* __restrict__ ideg) {
    int e = blockIdx.x * blockDim.x + threadIdx.x;
    if (e < N_EDGES) atomicAdd(&ideg[row[e]], 1);
}

__global__ void dis_kernel(const int* __restrict__ ideg, float* __restrict__ dis) {
    int i = blockIdx.x * blockDim.x + threadIdx.x;
    if (i < N_NODES) {
        int d = ideg[i];
        dis[i] = d > 0 ? rsqrtf((float)d) : 0.0f;
    }
}

__global__ void w_kernel(const int* __restrict__ row, const int* __restrict__ col,
                         const float* __restrict__ dis, float* __restrict__ w) {
    int e = blockIdx.x * blockDim.x + threadIdx.x;
    if (e < N_EDGES) w[e] = dis[row[e]] * dis[col[e]];
}

// One-workgroup exclusive prefix scan of ideg -> rowptr (N=100000 is tiny).
__global__ __launch_bounds__(1024)
void scan_kernel(const int* __restrict__ ideg, int* __restrict__ rowptr) {
    __shared__ int sums[1024];
    const int t = threadIdx.x;
    const int CHUNK = (N_NODES + 1023) / 1024;     // 98
    const int begin = t * CHUNK;
    const int end   = begin + CHUNK < N_NODES ? begin + CHUNK : N_NODES;
    int s = 0;
    for (int i = begin; i < end; ++i) s += ideg[i];
    sums[t] = s;
    __syncthreads();
    for (int off = 1; off < 1024; off <<= 1) {     // Hillis-Steele inclusive scan
        int v = (t >= off) ? sums[t - off] : 0;
        __syncthreads();
        sums[t] += v;
        __syncthreads();
    }
    int excl = (t == 0) ? 0 : sums[t - 1];
    for (int i = begin; i < end; ++i) {
        rowptr[i] = excl;
        excl += ideg[i];
    }
    if (t == 1023) rowptr[N_NODES] = excl;         // == N_EDGES
}

__global__ void cursor_init_kernel(const int* __restrict__ rowptr, int* __restrict__ cur) {
    int i = blockIdx.x * blockDim.x + threadIdx.x;
    if (i < N_NODES) cur[i] = rowptr[i];
}

// Permute (col, w) into CSR order keyed by destination row.
__global__ void fill_kernel(const int* __restrict__ row, const int* __restrict__ col,
                            const float* __restrict__ w, int* __restrict__ cursor,
                            int* __restrict__ ecol, float* __restrict__ ew) {
    int e = blockIdx.x * blockDim.x + threadIdx.x;
    if (e < N_EDGES) {
        int pos = atomicAdd(&cursor[row[e]], 1);
        ecol[pos] = col[e];
        ew[pos]   = w[e];
    }
}

// ---------------------------------------------------------------------------
// Monomial coefficients of p(t) = sum_i C(K,i)/2^K * relu(temp[i]) (1-t)^i (1+t)^{K-i}
// Enables Horner: 10 SpMMs instead of 65.
// ---------------------------------------------------------------------------
__global__ void coef_kernel(const float* __restrict__ temp, float* __restrict__ coef) {
    if (threadIdx.x != 0 || blockIdx.x != 0) return;
    const double binom[K_ORDER + 1] = {1, 10, 45, 120, 210, 252, 210, 120, 45, 10, 1};
    double a[K_ORDER + 1];
    for (int j = 0; j <= K_ORDER; ++j) a[j] = 0.0;
    for (int i = 0; i <= K_ORDER; ++i) {
        double p[K_ORDER + 1];
        for (int j = 0; j <= K_ORDER; ++j) p[j] = 0.0;
        p[0] = 1.0;
        int deg = 0;
        for (int m = 0; m < i; ++m) {            // multiply by (1 - t)
            for (int j = deg + 1; j >= 1; --j) p[j] = p[j] - p[j - 1];
            ++deg;
        }
        for (int m = 0; m < K_ORDER - i; ++m) {  // multiply by (1 + t)
            for (int j = deg + 1; j >= 1; --j) p[j] = p[j] + p[j - 1];
            ++deg;
        }
        float ti = temp[i];
        double c = (binom[i] / 1024.0) * (ti > 0.0f ? (double)ti : 0.0);
        for (int j = 0; j <= K_ORDER; ++j) a[j] += c * p[j];
    }
    for (int j = 0; j <= K_ORDER; ++j) coef[j] = (float)a[j];
}

// r = coef[K] * h   (Horner seed)
__global__ void scale_init_kernel(const float* __restrict__ h, float* __restrict__ out,
                                  const float* __restrict__ coef, int j) {
    int i = blockIdx.x * blockDim.x + threadIdx.x;
    if (i < N_NODES * CLS) out[i] = coef[j] * h[i];
}

// Fused Horner step: vout[n] = coef[j]*h[n] + sum_{p in row n} ew[p]*vin[ecol[p]]
// One wave32 per node; lane owns channels (lane, lane+32). Gather-only, no atomics.
__global__ __launch_bounds__(256)
void spmm_gather_kernel(const int* __restrict__ rowptr, const int* __restrict__ ecol,
                        const float* __restrict__ ew,
                        const float* __restrict__ h, const float* __restrict__ coef, int j,
                        const float* __restrict__ vin, float* __restrict__ vout) {
    const int lane = threadIdx.x & 31;
    const int wave = threadIdx.x >> 5;
    const int node = blockIdx.x * 8 + wave;
    if (node >= N_NODES) return;
    const float cj = coef[j];
    float acc0 = cj * h[(size_t)node * CLS + lane];
    float acc1 = cj * h[(size_t)node * CLS + lane + 32];
    const int pbeg = rowptr[node];
    const int pend = rowptr[node + 1];
    int p = pbeg;
    // 2-edge unrolled main loop for memory-level parallelism
    for (; p + 1 < pend; p += 2) {
        const int   c0 = ecol[p],     c1 = ecol[p + 1];
        const float w0 = ew[p],       w1 = ew[p + 1];
        const float* v0 = vin + (size_t)c0 * CLS;
        const float* v1 = vin + (size_t)c1 * CLS;
        acc0 += w0 * v0[lane];
        acc1 += w0 * v0[lane + 32];
        acc0 += w1 * v1[lane];
        acc1 += w1 * v1[lane + 32];
    }
    if (p < pend) {
        const int   c0 = ecol[p];
        const float w0 = ew[p];
        const float* v0 = vin + (size_t)c0 * CLS;
        acc0 += w0 * v0[lane];
        acc1 += w0 * v0[lane + 32];
    }
    vout[(size_t)node * CLS + lane]      = acc0;
    vout[(size_t)node * CLS + lane + 32] = acc1;
}

// log-softmax over 64 classes; one wave32 per node, 2 classes per lane.
__global__ __launch_bounds__(256)
void log_softmax_kernel(const float* __restrict__ v, float* __restrict__ out) {
    const int lane = threadIdx.x & 31;
    const int wave = threadIdx.x >> 5;
    const int node = blockIdx.x * 8 + wave;
    if (node >= N_NODES) return;
    const float* p = v + (size_t)node * CLS;
    float a = p[lane], b = p[lane + 32];
    float m = fmaxf(a, b);
#pragma unroll
    for (int off = 16; off >= 1; off >>= 1)
        m = fmaxf(m, __shfl_xor(m, off, 32));
    float s = expf(a - m) + expf(b - m);
#pragma unroll
    for (int off = 16; off >= 1; off >>= 1)
        s += __shfl_xor(s, off, 32);
    const float ls = logf(s);
    out[(size_t)node * CLS + lane]      = a - m - ls;
    out[(size_t)node * CLS + lane + 32] = b - m - ls;
}

// ---------------------------------------------------------------------------
extern "C" void kernel_launch(void* const* d_in, const int* in_sizes, int n_in,
                              void* d_out, int out_size, void* d_ws, size_t ws_size,
                              hipStream_t stream) {
    const float* x    = (const float*)d_in[0];
    const int*   edge = (const int*)d_in[1];
    const float* W1   = (const float*)d_in[2];
    const float* b1   = (const float*)d_in[3];
    const float* W2   = (const float*)d_in[4];
    const float* b2   = (const float*)d_in[5];
    const float* temp = (const float*)d_in[6];
    const int* row = edge;             // edge_index[0]
    const int* col = edge + N_EDGES;   // edge_index[1]

    char* ws = (char*)d_ws;
    const size_t NC = (size_t)N_NODES * CLS * sizeof(float);   // 25.6 MB
    const size_t NB = (size_t)(N_NODES + 4) * sizeof(int);
    const size_t EB = (size_t)N_EDGES * sizeof(float);
    size_t off = 0;
    float* h      = (float*)(ws + off); off += NC;
    float* r0     = (float*)(ws + off); off += NC;
    float* r1     = (float*)(ws + off); off += NC;
    int*   ideg   = (int*)  (ws + off); off += NB;
    float* dis    = (float*)(ws + off); off += NB;
    int*   rowptr = (int*)  (ws + off); off += NB;
    int*   cursor = (int*)  (ws + off); off += NB;
    float* wbuf   = (float*)(ws + off); off += EB;
    int*   ecol   = (int*)  (ws + off); off += EB;
    float* ewght  = (float*)(ws + off); off += EB;
    float* coef   = (float*)(ws + off); off += 64;

    const int nblk = (N_NODES + 255) / 256;
    const int eblk = (N_EDGES + 255) / 256;

    // 1) MLP with f32 WMMA
    mlp_wmma_kernel<<<(N_NODES + 127) / 128, 256, 0, stream>>>(x, W1, b1, W2, b2, h);

    // 2) degrees, sym-normalized edge weights
    izero_kernel<<<nblk, 256, 0, stream>>>(ideg, N_NODES);
    count_kernel<<<eblk, 256, 0, stream>>>(row, ideg);
    dis_kernel<<<nblk, 256, 0, stream>>>(ideg, dis);
    w_kernel<<<eblk, 256, 0, stream>>>(row, col, dis, wbuf);

    // 3) CSR-by-destination build (once) -> gather-only SpMM, no per-step atomics
    scan_kernel<<<1, 1024, 0, stream>>>(ideg, rowptr);
    cursor_init_kernel<<<nblk, 256, 0, stream>>>(rowptr, cursor);
    fill_kernel<<<eblk, 256, 0, stream>>>(row, col, wbuf, cursor, ecol, ewght);

    // 4) polynomial monomial coefficients (Bernstein -> monomial)
    coef_kernel<<<1, 32, 0, stream>>>(temp, coef);

    // 5) Horner: r = a_K h; for j=K-1..0: r = A r + a_j h   (10 gather SpMMs)
    const int nc_blocks = (N_NODES * CLS + 255) / 256;
    const int nd_blocks = (N_NODES + 7) / 8;
    float* cur = r0;
    float* nxt = r1;
    scale_init_kernel<<<nc_blocks, 256, 0, stream>>>(h, cur, coef, K_ORDER);
    for (int j = K_ORDER - 1; j >= 0; --j) {
        spmm_gather_kernel<<<nd_blocks, 256, 0, stream>>>(rowptr, ecol, ewght,
                                                          h, coef, j, cur, nxt);
        float* tmp = cur; cur = nxt; nxt = tmp;
    }

    // 6) log-softmax -> d_out
    log_softmax_kernel<<<nd_blocks, 256, 0, stream>>>(cur, (float*)d_out);
}